// EnhancedGNN_7748121002089
// MI455X (gfx1250) — compile-verified
//
#include <hip/hip_runtime.h>
#include <hip/hip_bf16.h>

typedef __attribute__((ext_vector_type(2))) float v2f;
typedef __attribute__((ext_vector_type(8))) float v8f;

#define DIVUP(a, b) (((a) + (b) - 1) / (b))

// ---------------- degree / norm precompute ----------------

__global__ void k_init_deg(float* __restrict__ deg, int n) {
    int i = blockIdx.x * blockDim.x + threadIdx.x;
    if (i < n) deg[i] = 1.0f;  // self-loop weight
}

__global__ void k_accum_deg(const int* __restrict__ dst, const float* __restrict__ w,
                            float* __restrict__ deg, int ne) {
    int e = blockIdx.x * blockDim.x + threadIdx.x;
    if (e < ne) atomicAdd(&deg[dst[e]], w[e]);
}

__global__ void k_dinv(float* __restrict__ deg, int n) {
    int i = blockIdx.x * blockDim.x + threadIdx.x;
    if (i < n) {
        float d = deg[i];
        deg[i] = (d > 0.0f) ? rsqrtf(fmaxf(d, 1e-30f)) : 0.0f;
    }
}

__global__ void k_norm(const int* __restrict__ src, const int* __restrict__ dst,
                       const float* __restrict__ w, const float* __restrict__ dinv,
                       float* __restrict__ nrm, int ne) {
    int e = blockIdx.x * blockDim.x + threadIdx.x;
    if (e < ne) nrm[e] = dinv[src[e]] * w[e] * dinv[dst[e]];
}

// ---------------- dense GEMM via V_WMMA_F32_16X16X4_F32 ----------------
// out[M x N] = A[M x K] @ W[K x N], row-major, fp32. One wave per 16-row tile,
// covering all N/16 column tiles (A fragment reused). M must be handled with
// clamped loads / guarded stores (M = 100000 is a multiple of 16 here).
template <int K, int N>
__global__ __launch_bounds__(256) void k_gemm(const float* __restrict__ A,
                                              const float* __restrict__ W,
                                              float* __restrict__ out, int M) {
    constexpr int NT = N / 16;
    const int lane = threadIdx.x & 31;
    const int wave = blockIdx.x * (blockDim.x >> 5) + (threadIdx.x >> 5);
    const int m0 = wave * 16;
    if (m0 >= M) return;

    const int hi = lane >> 4;   // 0 | 1
    const int lm = lane & 15;   // 0..15

    v8f acc[NT] = {};

    int arow = m0 + lm;
    if (arow >= M) arow = M - 1;               // clamp (padding rows unused)
    const float* arp = A + (size_t)arow * K;

    for (int k0 = 0; k0 < K; k0 += 4) {
        const int ka = k0 + hi * 2;
        v2f a;
        a.x = arp[ka];
        a.y = arp[ka + 1];
#pragma unroll
        for (int t = 0; t < NT; ++t) {
            v2f b;
            b.x = W[(size_t)ka * N + t * 16 + lm];
            b.y = W[(size_t)(ka + 1) * N + t * 16 + lm];
            acc[t] = __builtin_amdgcn_wmma_f32_16x16x4_f32(
                false, a, false, b, (short)0, acc[t], false, false);
        }
    }

#pragma unroll
    for (int t = 0; t < NT; ++t) {
#pragma unroll
        for (int r = 0; r < 8; ++r) {
            int m = m0 + r + hi * 8;
            if (m < M) out[(size_t)m * N + t * 16 + lm] = acc[t][r];
        }
    }
}

// ---------------- aggregation ----------------
// agg[v] = dinv[v]^2 * xw[v]  (self loop, also initializes agg)
template <int N>
__global__ void k_selfloop(const float* __restrict__ dinv, const float* __restrict__ xw,
                           float* __restrict__ agg, int n) {
    constexpr int Q = N / 4;
    unsigned tid = blockIdx.x * blockDim.x + threadIdx.x;
    unsigned v = tid / Q, q = tid % Q;
    if (v >= (unsigned)n) return;
    float di = dinv[v];
    float c = di * di;
    const float4 x = *(const float4*)(xw + (size_t)v * N + q * 4);
    float4 r;
    r.x = c * x.x; r.y = c * x.y; r.z = c * x.z; r.w = c * x.w;
    *(float4*)(agg + (size_t)v * N + q * 4) = r;
}

// agg[dst[e]] += norm[e] * xw[src[e]]  — float4 per thread, coalesced rows,
// native fp32 global atomics (bandwidth-bound stage).
template <int N>
__global__ void k_scatter(const int* __restrict__ src, const int* __restrict__ dst,
                          const float* __restrict__ nrm, const float* __restrict__ xw,
                          float* __restrict__ agg, int ne) {
    constexpr int Q = N / 4;
    unsigned tid = blockIdx.x * blockDim.x + threadIdx.x;
    unsigned e = tid / Q, q = tid % Q;
    if (e >= (unsigned)ne) return;
    float c = nrm[e];
    int s = src[e], d = dst[e];
    const float4 x = *(const float4*)(xw + (size_t)s * N + q * 4);
    float* p = agg + (size_t)d * N + q * 4;
    atomicAdd(p + 0, c * x.x);
    atomicAdd(p + 1, c * x.y);
    atomicAdd(p + 2, c * x.z);
    atomicAdd(p + 3, c * x.w);
}

template <int N>
__global__ void k_bias_relu(const float* __restrict__ agg, const float* __restrict__ b,
                            float* __restrict__ h, int n) {
    unsigned tid = blockIdx.x * blockDim.x + threadIdx.x;
    if (tid >= (unsigned)n * N) return;
    int f = tid % N;
    h[tid] = fmaxf(agg[tid] + b[f], 0.0f);
}

// ---------------- pooling + FC ----------------

__global__ void k_zero(float* __restrict__ p, int n) {
    int i = blockIdx.x * blockDim.x + threadIdx.x;
    if (i < n) p[i] = 0.0f;
}

__global__ void k_cnt(const int* __restrict__ batch, float* __restrict__ cnt, int n) {
    int i = blockIdx.x * blockDim.x + threadIdx.x;
    if (i < n) atomicAdd(&cnt[batch[i]], 1.0f);
}

__global__ void k_pool(const int* __restrict__ batch, const float* __restrict__ h,
                       float* __restrict__ pooled, int n) {
    unsigned tid = blockIdx.x * blockDim.x + threadIdx.x;
    unsigned v = tid / 16, q = tid % 16;   // 64 feats / 4 per thread
    if (v >= (unsigned)n) return;
    int g = batch[v];
    const float4 x = *(const float4*)(h + (size_t)v * 64 + q * 4);
    float* p = pooled + (size_t)g * 64 + q * 4;
    atomicAdd(p + 0, x.x);
    atomicAdd(p + 1, x.y);
    atomicAdd(p + 2, x.z);
    atomicAdd(p + 3, x.w);
}

__global__ void k_fc(const float* __restrict__ pooled, const float* __restrict__ cnt,
                     const float* __restrict__ Wfc, const float* __restrict__ bfc,
                     float* __restrict__ out) {
    int g = threadIdx.x;  // 64 graphs
    float s = 0.0f;
#pragma unroll
    for (int f = 0; f < 64; ++f) s += pooled[g * 64 + f] * Wfc[f];
    out[g] = s / fmaxf(cnt[g], 1.0f) + bfc[0];
}

// ---------------- launcher ----------------

extern "C" void kernel_launch(void* const* d_in, const int* in_sizes, int n_in,
                              void* d_out, int out_size, void* d_ws, size_t ws_size,
                              hipStream_t stream) {
    const float* x     = (const float*)d_in[0];
    const int*   ei    = (const int*)  d_in[1];   // [2, NE]
    const float* ew    = (const float*)d_in[2];
    const int*   batch = (const int*)  d_in[3];
    const float* W1 = (const float*)d_in[4];
    const float* b1 = (const float*)d_in[5];
    const float* W2 = (const float*)d_in[6];
    const float* b2 = (const float*)d_in[7];
    const float* W3 = (const float*)d_in[8];
    const float* b3 = (const float*)d_in[9];
    const float* Wfc = (const float*)d_in[10];
    const float* bfc = (const float*)d_in[11];
    float* out = (float*)d_out;

    const int n  = in_sizes[3];   // 100000 nodes
    const int ne = in_sizes[2];   // 1600000 edges
    const int* src = ei;
    const int* dst = ei + ne;

    // workspace layout (floats)
    float* ws = (float*)d_ws;
    const size_t BIG = (size_t)n * 128;
    float* bufA   = ws;             // [n,128]
    float* bufB   = bufA + BIG;     // [n,128]
    float* dinv   = bufB + BIG;     // [n]   (deg -> dinv in place)
    float* nrm    = dinv + n;       // [ne]
    float* pooled = nrm + ne;       // [64*64]
    float* cnt    = pooled + 64 * 64;  // [64]

    const int TB = 256;

    // gcn_norm
    k_init_deg<<<DIVUP(n, TB), TB, 0, stream>>>(dinv, n);
    k_accum_deg<<<DIVUP(ne, TB), TB, 0, stream>>>(dst, ew, dinv, ne);
    k_dinv<<<DIVUP(n, TB), TB, 0, stream>>>(dinv, n);
    k_norm<<<DIVUP(ne, TB), TB, 0, stream>>>(src, dst, ew, dinv, nrm, ne);

    const int mt = DIVUP(n, 16);          // 6250 row tiles
    const int gblk = DIVUP(mt, 8);        // 8 waves / block

    // layer 1: [n,16] @ [16,64]
    k_gemm<16, 64><<<gblk, TB, 0, stream>>>(x, W1, bufA, n);
    k_selfloop<64><<<DIVUP(n * 16, TB), TB, 0, stream>>>(dinv, bufA, bufB, n);
    k_scatter<64><<<DIVUP(ne * 16, TB), TB, 0, stream>>>(src, dst, nrm, bufA, bufB, ne);
    k_bias_relu<64><<<DIVUP(n * 64, TB), TB, 0, stream>>>(bufB, b1, bufA, n);

    // layer 2: [n,64] @ [64,128]
    k_gemm<64, 128><<<gblk, TB, 0, stream>>>(bufA, W2, bufB, n);
    k_selfloop<128><<<DIVUP(n * 32, TB), TB, 0, stream>>>(dinv, bufB, bufA, n);
    k_scatter<128><<<DIVUP(ne * 32, TB), TB, 0, stream>>>(src, dst, nrm, bufB, bufA, ne);
    k_bias_relu<128><<<DIVUP(n * 128, TB), TB, 0, stream>>>(bufA, b2, bufB, n);

    // layer 3: [n,128] @ [128,64]
    k_gemm<128, 64><<<gblk, TB, 0, stream>>>(bufB, W3, bufA, n);
    k_selfloop<64><<<DIVUP(n * 16, TB), TB, 0, stream>>>(dinv, bufA, bufB, n);
    k_scatter<64><<<DIVUP(ne * 16, TB), TB, 0, stream>>>(src, dst, nrm, bufA, bufB, ne);
    k_bias_relu<64><<<DIVUP(n * 64, TB), TB, 0, stream>>>(bufB, b3, bufA, n);

    // mean pool over graphs + FC
    k_zero<<<DIVUP(64 * 64 + 64, TB), TB, 0, stream>>>(pooled, 64 * 64 + 64);
    k_cnt<<<DIVUP(n, TB), TB, 0, stream>>>(batch, cnt, n);
    k_pool<<<DIVUP(n * 16, TB), TB, 0, stream>>>(batch, bufA, pooled, n);
    k_fc<<<1, 64, 0, stream>>>(pooled, cnt, Wfc, bfc, out);
}